// NaNRandomDense_43550968381955
// MI455X (gfx1250) — compile-verified
//
#include <hip/hip_runtime.h>
#include <hip/hip_bf16.h>
#include <stdint.h>

#define BATCH 131072
#define SIZE  1024
#define WAVES_PER_BLOCK 8
#define THREADS (WAVES_PER_BLOCK * 32)

__device__ __forceinline__ bool finitef(float x) {
    // finite <=> exponent field != all ones (excludes NaN and +/-Inf), exact bit test
    return (__float_as_uint(x) & 0x7F800000u) != 0x7F800000u;
}

__global__ __launch_bounds__(THREADS)
void NaNRandomDense_kernel(const float* __restrict__ X,
                           const float* __restrict__ w,
                           const float* __restrict__ b,
                           float* __restrict__ out) {
    __shared__ float smem[WAVES_PER_BLOCK * SIZE];   // 32 KB / block

    const int lane = threadIdx.x & 31;
    const int wave = threadIdx.x >> 5;
    const int row  = blockIdx.x * WAVES_PER_BLOCK + wave;   // exact cover: 16384*8

    float* buf = &smem[wave * SIZE];
    // Low 32 bits of the flat shared address are the LDS byte address (ISA 10.2).
    const uint32_t lds_base = (uint32_t)(uintptr_t)buf;

    // ---- Phase A: async DMA the 4KB row into LDS, coalesced b128 per lane ----
    const uint32_t row_byte = (uint32_t)row * (SIZE * 4u);   // <= 2^29, fits signed i32
    #pragma unroll
    for (int t = 0; t < 8; ++t) {
        uint32_t eb = (uint32_t)((t * 128 + lane * 4) * 4);  // byte offset within row
        uint32_t va = row_byte + eb;                          // vaddr (i32 offset, GVS)
        uint32_t la = lds_base + eb;                          // LDS dest address
        asm volatile("global_load_async_to_lds_b128 %0, %1, %2"
                     :: "v"(la), "v"(va), "s"(X)
                     : "memory");
    }
    asm volatile("s_wait_asynccnt 0x0" ::: "memory");
    __builtin_amdgcn_wave_barrier();

    // ---- Phase B: each lane reads its contiguous 32-element chunk (stable order) ----
    float4 v[8];
    const float4* bv4 = (const float4*)buf;
    #pragma unroll
    for (int j = 0; j < 8; ++j) v[j] = bv4[lane * 8 + j];    // elems lane*32 + 4j ..

    const float* f = (const float*)v;
    int cnt = 0;
    #pragma unroll
    for (int i = 0; i < 32; ++i) cnt += finitef(f[i]) ? 1 : 0;

    // wave32 inclusive scan of per-lane finite counts
    int incl = cnt;
    #pragma unroll
    for (int d = 1; d < 32; d <<= 1) {
        int n = __shfl_up(incl, d, 32);
        if (lane >= d) incl += n;
    }
    const int base  = incl - cnt;              // exclusive prefix = dest start for this lane
    const int total = __shfl(incl, 31, 32);    // r = finite count of the whole row

    __builtin_amdgcn_wave_barrier();

    // ---- Phase C: scatter survivors back into the SAME LDS buffer, compacted ----
    // Safe: whole row already lives in registers; DS ops of one wave are in-order.
    int pos = base;
    #pragma unroll
    for (int i = 0; i < 32; ++i) {
        float x = f[i];
        bool fin = finitef(x);
        if (fin) buf[pos] = x;
        pos += fin ? 1 : 0;
    }

    __builtin_amdgcn_wave_barrier();

    // ---- Phase D: coalesced readback, affine + zero-tail mask, b128 stores ----
    const float4* wv = (const float4*)w;
    const float4* bv = (const float4*)b;
    float4* orow = (float4*)(out + (size_t)row * SIZE);
    #pragma unroll
    for (int t = 0; t < 8; ++t) {
        int e = t * 128 + lane * 4;            // output position
        float4 xc = bv4[e >> 2];
        float4 ww = wv[e >> 2];
        float4 bb = bv[e >> 2];
        float4 o;
        o.x = (e + 0 < total) ? xc.x * ww.x + bb.x : 0.0f;
        o.y = (e + 1 < total) ? xc.y * ww.y + bb.y : 0.0f;
        o.z = (e + 2 < total) ? xc.z * ww.z + bb.z : 0.0f;
        o.w = (e + 3 < total) ? xc.w * ww.w + bb.w : 0.0f;
        orow[e >> 2] = o;                      // every output written -> poison overwritten
    }
}

extern "C" void kernel_launch(void* const* d_in, const int* in_sizes, int n_in,
                              void* d_out, int out_size, void* d_ws, size_t ws_size,
                              hipStream_t stream) {
    const float* X = (const float*)d_in[0];   // [131072, 1024] f32 (with NaNs)
    const float* w = (const float*)d_in[1];   // [1024] f32
    const float* b = (const float*)d_in[2];   // [1024] f32
    float* out     = (float*)d_out;           // [131072, 1024] f32

    dim3 grid(BATCH / WAVES_PER_BLOCK);       // 16384 blocks x 256 threads = 1 wave/row
    NaNRandomDense_kernel<<<grid, THREADS, 0, stream>>>(X, w, b, out);
}